// Distmult_41858751266871
// MI455X (gfx1250) — compile-verified
//
#include <hip/hip_runtime.h>
#include <hip/hip_bf16.h>
#include <stdint.h>

#define DIMS        256
#define TILE_ROWS   16
#define STAGE_COLS  64
#define NSTAGES     (DIMS / STAGE_COLS)          // 4
#define PITCH       68                           // 64 + 4 pad floats -> LDS bank stride 4
#define BUF_FLOATS  (TILE_ROWS * PITCH)          // 1088
#define BUF_BYTES   (BUF_FLOATS * 4)             // 4352
#define WAVE_FLOATS (6 * BUF_FLOATS)             // 3 arrays x 2 buffers = 6528 floats
#define WAVES_PER_BLOCK 2

typedef uint32_t u32;
typedef uint64_t u64;
typedef __attribute__((ext_vector_type(2))) float v2f;
typedef __attribute__((ext_vector_type(8))) float v8f;
typedef __attribute__((ext_vector_type(4))) u32   v4u;
typedef __attribute__((ext_vector_type(8))) int   v8i;
typedef __attribute__((ext_vector_type(4))) int   v4i;

// ---- Tensor DMA descriptor builders (CDNA5 ISA) ----

// Group 0: count=1, lds_addr, 57-bit global addr, type=2("image"), optional gather.
__device__ __forceinline__ v4u tdm_g0(u32 lds_addr, const float* g, u32 gather) {
    u64 ga = (u64)(uintptr_t)g;
    v4u g0;
    g0.x = 1u | (gather << 31);                         // count=1; gather_mode; 16-bit idx
    g0.y = lds_addr;                                    // LDS byte address
    g0.z = (u32)ga;                                     // global_addr[31:0]
    g0.w = (u32)((ga >> 32) & 0x01FFFFFFu) | (2u << 30); // global_addr[56:32], type=2
    return g0;
}

// Group 1: data_size=4B, pad 4 DWORDs every 64 DWORDs (-> 68-float LDS pitch,
// bank-conflict-free WMMA-layout ds_load_b64), 16x64 tile, row stride = DIMS.
__device__ __forceinline__ v8i tdm_g1(u32 tdim0, u32 tdim1) {
    v8i g1;
    g1[0] = (int)((2u << 16)      // data_size = 4 bytes
                | (1u << 20)      // pad_enable
                | (5u << 22)      // pad_interval: 2^(5+1) = 64 DWORDs
                | (3u << 25));    // pad_amount:   3+1    = 4 DWORDs
    g1[1] = (int)((tdim0 & 0xFFFFu) << 16);                                   // tensor_dim0 lo16
    g1[2] = (int)(((tdim0 >> 16) & 0xFFFFu) | ((tdim1 & 0xFFFFu) << 16));     // dim0 hi | dim1 lo
    g1[3] = (int)(((tdim1 >> 16) & 0xFFFFu) | ((u32)STAGE_COLS << 16));       // dim1 hi | tile_dim0
    g1[4] = (int)TILE_ROWS;       // tile_dim1 (= #gather indices in gather mode); tile_dim2=0
    g1[5] = (int)DIMS;            // tensor_dim0_stride lo32 (elements)
    g1[6] = 0;                    // stride hi16 | dim1_stride lo16
    g1[7] = 0;
    return g1;
}

// Dense 16x64 tile load (head/tail).
__device__ __forceinline__ void tdm_load_tile(u32 lds_addr, const float* g) {
    v4i z4 = {0, 0, 0, 0};
    v8i z8 = {0, 0, 0, 0, 0, 0, 0, 0};
    __builtin_amdgcn_tensor_load_to_lds(tdm_g0(lds_addr, g, 0u),
                                        tdm_g1((u32)DIMS, (u32)TILE_ROWS),
                                        z4, z4, z8, 0);
}

// Gathered 16-row x 64-col load (rel_embeds rows selected by 16-bit indices).
__device__ __forceinline__ void tdm_load_gather(u32 lds_addr, const float* g, const u32* pk) {
    v4i g2 = {(int)pk[0], (int)pk[1], (int)pk[2], (int)pk[3]};
    v4i g3 = {(int)pk[4], (int)pk[5], (int)pk[6], (int)pk[7]};
    v8i z8 = {0, 0, 0, 0, 0, 0, 0, 0};
    __builtin_amdgcn_tensor_load_to_lds(tdm_g0(lds_addr, g, 1u),
                                        tdm_g1((u32)DIMS, 512u),
                                        g2, g3, z8, 0);
}

__device__ __forceinline__ float diag_pick(v8f c, int i) {
    float s = c[0];
    s = (i == 1) ? c[1] : s;
    s = (i == 2) ? c[2] : s;
    s = (i == 3) ? c[3] : s;
    s = (i == 4) ? c[4] : s;
    s = (i == 5) ? c[5] : s;
    s = (i == 6) ? c[6] : s;
    s = (i == 7) ? c[7] : s;
    return s;
}

// One wave32 per 16-row tile.  TDM stages 16x64 f32 chunks of head/tail (dense)
// and rel (index-gather) into padded LDS; ds_load_b64 reads them directly in the
// f32 WMMA 16x16x4 operand layout (bank-conflict-free at pitch 68); 64 chained
// v_wmma_f32_16x16x4_f32 produce all 16 scores on the accumulator diagonal.
__global__ __launch_bounds__(WAVES_PER_BLOCK * 32) void distmult_tdm_wmma_kernel(
    const float* __restrict__ head,
    const float* __restrict__ tail,
    const int*   __restrict__ rel_ids,
    const float* __restrict__ lower_bound,
    const float* __restrict__ upper_bound,
    const float* __restrict__ rel,
    float* __restrict__ out,
    int n_pos, int n_rows)
{
    __shared__ float smem[WAVES_PER_BLOCK * WAVE_FLOATS];   // 52,224 bytes

    const int lane = threadIdx.x & 31;
    const int wib  = __builtin_amdgcn_readfirstlane(threadIdx.x >> 5);
    const int R    = __builtin_amdgcn_readfirstlane((blockIdx.x * WAVES_PER_BLOCK + wib) * TILE_ROWS);
    if (R >= n_rows) return;                                 // uniform per wave

    float* my = smem + wib * WAVE_FLOATS;
    const u32 myB = (u32)(uintptr_t)my;                      // LDS byte base (flat low bits)

    // Uniform scalar loads of the 16 relation ids; pack as 16-bit gather indices.
    const int* idp = rel_ids + R;
    u32 pk[8];
#pragma unroll
    for (int i = 0; i < 8; ++i)
        pk[i] = ((u32)idp[2 * i] & 0xFFFFu) | (((u32)idp[2 * i + 1] & 0xFFFFu) << 16);

    const int r    = lane & 15;               // M (A) / N (B)
    const int colL = (lane >> 4) << 1;        // 0 for lanes 0-15, 2 for lanes 16-31
    const int ldsRow = r * PITCH + colL;      // banks 4r+{0,1} / 4r+{2,3} -> all 64 banks

    const float* gH = head + (size_t)R * DIMS;
    const float* gT = tail + (size_t)R * DIMS;

    v8f acc = {};

    // --- double-buffered TDM pipeline over 4 K-stages of 64 columns ---
#define ISSUE_STAGE(s, b)                                                     \
    do {                                                                      \
        tdm_load_tile(myB + (0 * 2 + (b)) * BUF_BYTES, gH + (s) * STAGE_COLS);\
        tdm_load_tile(myB + (1 * 2 + (b)) * BUF_BYTES, gT + (s) * STAGE_COLS);\
        tdm_load_gather(myB + (2 * 2 + (b)) * BUF_BYTES, rel + (s) * STAGE_COLS, pk); \
    } while (0)

#define COMPUTE_STAGE(b)                                                      \
    do {                                                                      \
        const float* H = my + (0 * 2 + (b)) * BUF_FLOATS + ldsRow;            \
        const float* T = my + (1 * 2 + (b)) * BUF_FLOATS + ldsRow;            \
        const float* Rm = my + (2 * 2 + (b)) * BUF_FLOATS + ldsRow;           \
        _Pragma("unroll")                                                     \
        for (int c = 0; c < STAGE_COLS / 4; ++c) {                            \
            v2f h  = *(const v2f*)(H + c * 4);   /* ds_load_b64 */            \
            v2f t  = *(const v2f*)(T + c * 4);                                \
            v2f rv = *(const v2f*)(Rm + c * 4);                               \
            v2f a  = h * t;                      /* v_pk_mul_f32 */           \
            acc = __builtin_amdgcn_wmma_f32_16x16x4_f32(                      \
                false, a, false, rv, (short)0, acc, false, false);            \
        }                                                                     \
    } while (0)

    ISSUE_STAGE(0, 0);
    ISSUE_STAGE(1, 1);

    __builtin_amdgcn_s_wait_tensorcnt(3);          // stage 0 landed (in-order)
    asm volatile("" ::: "memory");
    COMPUTE_STAGE(0);
    asm volatile("s_wait_dscnt 0" ::: "memory");   // buf0 reads done before overwrite
    ISSUE_STAGE(2, 0);

    __builtin_amdgcn_s_wait_tensorcnt(3);          // stage 1 landed
    asm volatile("" ::: "memory");
    COMPUTE_STAGE(1);
    asm volatile("s_wait_dscnt 0" ::: "memory");
    ISSUE_STAGE(3, 1);

    __builtin_amdgcn_s_wait_tensorcnt(3);          // stage 2 landed
    asm volatile("" ::: "memory");
    COMPUTE_STAGE(2);

    __builtin_amdgcn_s_wait_tensorcnt(0);          // stage 3 landed
    asm volatile("" ::: "memory");
    COMPUTE_STAGE(3);

#undef ISSUE_STAGE
#undef COMPUTE_STAGE

    // Diagonal of the 16x16 f32 accumulator:
    //   m = 0..7  -> lane m     , VGPR m
    //   m = 8..15 -> lane m + 16, VGPR m - 8
    int sel = -1;
    if (lane < 8)        sel = lane;
    else if (lane >= 24) sel = lane - 24;

    if (sel >= 0) {
        const int m   = (lane < 8) ? lane : (lane - 16);
        const int row = R + m;
        float s = diag_pick(acc, sel);
        if (row < n_pos) s = fminf(s, upper_bound[row]);
        else             s = fmaxf(s, lower_bound[row - n_pos]);
        out[row] = 1.0f / (1.0f + __expf(-s));
    }
}

extern "C" void kernel_launch(void* const* d_in, const int* in_sizes, int n_in,
                              void* d_out, int out_size, void* d_ws, size_t ws_size,
                              hipStream_t stream) {
    (void)n_in; (void)d_ws; (void)ws_size; (void)out_size;
    const float* head        = (const float*)d_in[0];
    const float* tail        = (const float*)d_in[1];
    const int*   rel_ids     = (const int*)  d_in[2];
    const float* lower_bound = (const float*)d_in[3];
    const float* upper_bound = (const float*)d_in[4];
    const float* rel_embeds  = (const float*)d_in[5];
    float* out = (float*)d_out;

    const int n_rows = in_sizes[2];   // B = 524288
    const int n_pos  = in_sizes[4];   // 131072

    const int tiles  = (n_rows + TILE_ROWS - 1) / TILE_ROWS;                 // 32768 waves
    const int blocks = (tiles + WAVES_PER_BLOCK - 1) / WAVES_PER_BLOCK;      // 16384

    distmult_tdm_wmma_kernel<<<dim3(blocks), dim3(WAVES_PER_BLOCK * 32), 0, stream>>>(
        head, tail, rel_ids, lower_bound, upper_bound, rel_embeds,
        out, n_pos, n_rows);
}